// SPDTangentSpace_81140522156467
// MI455X (gfx1250) — compile-verified
//
#include <hip/hip_runtime.h>
#include <math.h>

// ---------------------------------------------------------------------------
// SPD log-map, matmul-only formulation for CDNA5 (gfx1250) WMMA.
//   logm(A) = 2^K * log( (A/c)^(1/2^K) ) + log(c) I,  c = ||A||_F
//   sqrt via coupled inverse-free Newton-Schulz (pure 64x64 f32 GEMMs)
//   log(I+X) via Horner polynomial (pure GEMMs)
// One workgroup (8 wave32) per matrix; 4 x 16KB LDS ping-pong buffers.
// Each wave owns two C-tiles with the SAME tN -> B fragments shared.
// ---------------------------------------------------------------------------

typedef __attribute__((ext_vector_type(2))) float v2f;
typedef __attribute__((ext_vector_type(8))) float v8f;
typedef __attribute__((ext_vector_type(4))) int   v4i;

#define MAT_N     64
#define MAT_NN    (MAT_N * MAT_N)
#define TRI_N     2080            // 64*65/2
#define THREADS   256
#define K_SQRT    5
#define NS_ITERS  16
#define LOG_TERMS 8

// D = scale * (A @ B) + diag * I      (entry barrier only; no exit barrier)
// All operands are LDS-resident 64x64 f32, row-major stride 64.
// A is read TRANSPOSED via symmetry (all iterates are polynomials in the SPD
// input, hence symmetric) -> both fragment loads are lane-contiguous in LDS.
__device__ __forceinline__ void mm64_sym(const float* A, const float* B,
                                         float* D, float scale, float diag) {
  const int tid  = threadIdx.x;
  const int wave = tid >> 5;
  const int lane = tid & 31;
  const int l16  = lane & 15;
  const int kh   = (lane >> 4) << 1;      // K half: 0 or 2 (ISA A/B f32 layout)
  const int tN   = wave & 3;              // shared by both owned tiles
  const int tM0  = wave >> 2;             // 0..1 ; second tile is tM0+2
  const int mg0  = (tM0 << 4) + l16;
  const int mg1  = mg0 + 32;
  const int ng   = (tN << 4) + l16;
  __syncthreads();                        // operands visible to all waves
  v8f acc0 = {0.f, 0.f, 0.f, 0.f, 0.f, 0.f, 0.f, 0.f};
  v8f acc1 = {0.f, 0.f, 0.f, 0.f, 0.f, 0.f, 0.f, 0.f};
#pragma unroll
  for (int chunk = 0; chunk < 2; ++chunk) {
    // Preload 8 k-steps of fragments, then fire 16 back-to-back WMMAs:
    // one partial dscnt wait per chunk instead of a full drain per WMMA.
    v2f a0[8], a1[8], bb[8];
#pragma unroll
    for (int kk = 0; kk < 8; ++kk) {
      const int kr = (chunk * 8 + kk) * 4 + kh;
      a0[kk].x = A[kr * MAT_N + mg0];       // (M=mg0, K=kr..kr+1) via symmetry
      a0[kk].y = A[(kr + 1) * MAT_N + mg0];
      a1[kk].x = A[kr * MAT_N + mg1];
      a1[kk].y = A[(kr + 1) * MAT_N + mg1];
      bb[kk].x = B[kr * MAT_N + ng];        // (K=kr..kr+1, N=ng)
      bb[kk].y = B[(kr + 1) * MAT_N + ng];
    }
#pragma unroll
    for (int kk = 0; kk < 8; ++kk) {
      acc0 = __builtin_amdgcn_wmma_f32_16x16x4_f32(
          false, a0[kk], false, bb[kk], (short)0, acc0, false, false);
      acc1 = __builtin_amdgcn_wmma_f32_16x16x4_f32(
          false, a1[kk], false, bb[kk], (short)0, acc1, false, false);
    }
  }
  // C/D layout: VGPR r -> M = rbase + r; lanes 16-31 carry M += 8
  const int rb0 = (tM0 << 4) + ((lane >> 4) << 3);
#pragma unroll
  for (int r = 0; r < 8; ++r) {
    const int row0 = rb0 + r;
    const int row1 = row0 + 32;
    float v0 = scale * acc0[r];
    float v1 = scale * acc1[r];
    if (row0 == ng) v0 += diag;
    if (row1 == ng) v1 += diag;
    D[row0 * MAT_N + ng] = v0;
    D[row1 * MAT_N + ng] = v1;
  }
}

__global__ __launch_bounds__(THREADS)
void spd_logm_kernel(const float* __restrict__ Ag, float* __restrict__ outg) {
  __shared__ float smem[4 * MAT_NN];      // 64 KB (WGP has 320 KB LDS)
  float* P0 = smem;
  float* P1 = smem + MAT_NN;
  float* P2 = smem + 2 * MAT_NN;
  float* P3 = smem + 3 * MAT_NN;
  const int tid = threadIdx.x;
  const int b   = blockIdx.x;
  const float* Ain = Ag + (size_t)b * MAT_NN;

  // ---- load A -> P0 (async DMA to LDS when available, else b128 copy) ----
  {
    const float4* src = (const float4*)Ain;
#if __has_builtin(__builtin_amdgcn_global_load_async_to_lds_b128)
#pragma unroll
    for (int i = 0; i < 4; ++i) {
      __builtin_amdgcn_global_load_async_to_lds_b128(
          (__attribute__((address_space(1))) v4i*)(src + i * THREADS + tid),
          (__attribute__((address_space(3))) v4i*)((float4*)P0 + i * THREADS + tid),
          0, 0);
    }
#if __has_builtin(__builtin_amdgcn_s_wait_asynccnt)
    __builtin_amdgcn_s_wait_asynccnt(0);
#else
    asm volatile("s_wait_asynccnt 0" ::: "memory");
#endif
#else
    float4* dst = (float4*)P0;
#pragma unroll
    for (int i = 0; i < 4; ++i)
      dst[i * THREADS + tid] = src[i * THREADS + tid];
#endif
  }
  __syncthreads();

  // ---- c = ||A||_F  (guarantees spectrum(A/c) in (0,1]) ----
  float s = 0.f;
#pragma unroll
  for (int i = 0; i < 16; ++i) {
    float v = P0[i * THREADS + tid];
    s += v * v;
  }
  P1[tid] = s;                            // P1 free until Z-init
  __syncthreads();
  for (int off = THREADS >> 1; off > 0; off >>= 1) {
    if (tid < off) P1[tid] += P1[tid + off];
    __syncthreads();
  }
  const float c = sqrtf(P1[0]);
  __syncthreads();                        // done reading scratch
  const float inv_c = 1.f / c;

  // ---- Y0 = A/c (P0),  Z0 = I (P1) ----
#pragma unroll
  for (int i = 0; i < 16; ++i) {
    const int idx = i * THREADS + tid;
    P0[idx] *= inv_c;
    P1[idx] = ((idx / MAT_N) == (idx % MAT_N)) ? 1.f : 0.f;
  }
  // next mm64's entry barrier publishes these writes

  float* Y = P0; float* Z = P1; float* T = P2; float* W = P3;

  // ---- K_SQRT repeated matrix square roots via Newton-Schulz ----
#pragma unroll 1
  for (int stage = 0; stage < K_SQRT; ++stage) {
    if (stage > 0) {                      // fresh Z = I for this stage
      __syncthreads();                    // prior mm64 readers of Z done
#pragma unroll
      for (int i = 0; i < 16; ++i) {
        const int idx = i * THREADS + tid;
        Z[idx] = ((idx / MAT_N) == (idx % MAT_N)) ? 1.f : 0.f;
      }
    }
#pragma unroll 1
    for (int it = 0; it < NS_ITERS; ++it) {
      mm64_sym(Z, Y, T, -1.f, 3.f);       // T    = 3I - Z*Y
      mm64_sym(Y, T, W, 0.5f, 0.f);       // Ynew = 0.5 * Y * T   (into W)
      mm64_sym(T, Z, Y, 0.5f, 0.f);       // Znew = 0.5 * T * Z   (into old Y)
      float* oY = Y; float* oZ = Z; float* oT = T; float* oW = W;
      Y = oW; Z = oY; T = oZ; W = oT;     // rotate buffers
    }
    // Y now holds sqrt of this stage's input; Z holds its inverse sqrt.
  }

  // ---- X = Y - I ; Q = c_{m-1} I ----
  float* X = Z; float* Q = T; float* Tmp = W;   // Z/T/W contents dead now
  __syncthreads();                        // last mm64's writes/readers done
  {
    const float cm = (((LOG_TERMS - 1) & 1) ? -1.f : 1.f) / (float)LOG_TERMS;
#pragma unroll
    for (int i = 0; i < 16; ++i) {
      const int idx = i * THREADS + tid;
      const bool dg = (idx / MAT_N) == (idx % MAT_N);
      X[idx] = Y[idx] - (dg ? 1.f : 0.f);
      Q[idx] = dg ? cm : 0.f;
    }
  }

  // ---- Horner: Q = sum_{j} (-1)^j X^j / (j+1) ----
#pragma unroll 1
  for (int j = LOG_TERMS - 2; j >= 0; --j) {
    const float cj = ((j & 1) ? -1.f : 1.f) / (float)(j + 1);
    mm64_sym(X, Q, Tmp, 1.f, cj);         // Q' = X*Q + cj*I
    float* t = Q; Q = Tmp; Tmp = t;
  }

  // ---- logm = 2^K * X*Q + log(c)*I  (into Y buffer, X no longer needs Y) ----
  mm64_sym(X, Q, Y, (float)(1 << K_SQRT), logf(c));
  __syncthreads();                        // publish logm before readout

  // ---- upper-triangle (row-major) vectorization ----
  const float* L = Y;
  for (int o = tid; o < TRI_N; o += THREADS) {
    int rem = o, r = 0;
    while (rem >= (MAT_N - r)) { rem -= (MAT_N - r); ++r; }
    outg[(size_t)b * TRI_N + o] = L[r * MAT_N + r + rem];
  }
}

extern "C" void kernel_launch(void* const* d_in, const int* in_sizes, int n_in,
                              void* d_out, int out_size, void* d_ws, size_t ws_size,
                              hipStream_t stream) {
  const float* A = (const float*)d_in[0];
  float* out = (float*)d_out;
  const int B = in_sizes[0] / MAT_NN;     // 8192
  spd_logm_kernel<<<dim3(B), dim3(THREADS), 0, stream>>>(A, out);
}